// ConstraintModule_71279277245021
// MI455X (gfx1250) — compile-verified
//
#include <hip/hip_runtime.h>
#include <math.h>

// Problem constants (reference: B=2048, P_ROWS=128, N=64, Q=8)
#define NB   2048
#define PR   128
#define NN   64
#define NQ   8

typedef __attribute__((ext_vector_type(2))) float v2f;
typedef __attribute__((ext_vector_type(8))) float v8f;

#if defined(__has_builtin)
#if __has_builtin(__builtin_amdgcn_wmma_f32_16x16x4_f32)
#define HAVE_WMMA_F32 1
#endif
#endif

__global__ __launch_bounds__(256) void constraint_kernel(
    const float* __restrict__ v,    // (B, N)
    const float* __restrict__ A1,   // (B, PR, N)
    const float* __restrict__ b1,   // (B, PR)
    const float* __restrict__ z0,   // (B, N)
    const float* __restrict__ P,    // (B, Q, N, N)
    const float* __restrict__ qv,   // (B, Q, N)
    const float* __restrict__ r,    // (B, Q)
    float* __restrict__ out)        // (B, N)
{
    const int b    = blockIdx.x;
    const int tid  = threadIdx.x;
    const int wave = tid >> 5;
    const int lane = tid & 31;

    __shared__ float s_y0[NN];    // y0 = z0
    __shared__ float s_rho[NN];   // v_bar
    __shared__ float s_red[16];   // [0..7] kappa_lin per wave, [8..15] kappa_qc per q
    __shared__ float s_scal[2];   // norm_v, alpha

    // ---------------- Phase A: norm(v), v_bar, y0 into LDS (wave 0) -------------
    if (wave == 0) {
        const float* vb = v + (size_t)b * NN;
        float v0 = vb[lane];
        float v1 = vb[lane + 32];
        float ss = v0 * v0 + v1 * v1;
        #pragma unroll
        for (int m = 16; m >= 1; m >>= 1) ss += __shfl_xor(ss, m, 32);
        float nv  = sqrtf(ss);
        float inv = 1.0f / nv;
        s_rho[lane]      = v0 * inv;
        s_rho[lane + 32] = v1 * inv;
        const float* zb = z0 + (size_t)b * NN;
        s_y0[lane]      = zb[lane];
        s_y0[lane + 32] = zb[lane + 32];
        if (lane == 0) s_scal[0] = nv;
    }
    __syncthreads();

    // ---------------- Phase B: linear constraints (A1), per-wave 16 rows --------
    {
        // Each lane owns 2 columns; dot with y0(=z0) and rho from LDS.
        const float z0a = s_y0[2 * lane], z0b = s_y0[2 * lane + 1];
        const float vba = s_rho[2 * lane], vbb = s_rho[2 * lane + 1];
        const float* A1w = A1 + (size_t)b * PR * NN + (size_t)wave * 16 * NN;
        // preload this wave's 16 b1 values into lanes 0..15
        float bval = (lane < 16) ? b1[(size_t)b * PR + wave * 16 + lane] : 0.0f;
        float kl = 0.0f;
        #pragma unroll 4
        for (int row = 0; row < 16; ++row) {
            float2 a = *(const float2*)(A1w + row * NN + 2 * lane);
            float sz = a.x * z0a + a.y * z0b;
            float sv = a.x * vba + a.y * vbb;
            #pragma unroll
            for (int m = 16; m >= 1; m >>= 1) {
                sz += __shfl_xor(sz, m, 32);
                sv += __shfl_xor(sv, m, 32);
            }
            float brow = __shfl(bval, row, 32);
            if (lane == 0) {
                float t = sv / (brow - sz);   // (A1 v_bar)/(b1 - A1 z0)
                kl = fmaxf(kl, t);            // relu folded in via init 0
            }
        }
        if (lane == 0) s_red[wave] = kl;
    }

    // ---------------- Phase C: quadratic constraints via WMMA (wave q) ----------
    {
        const int qi   = wave;
        const float* Pq = P + ((size_t)(b * NQ + qi)) * NN * NN;
        const float* qq = qv + (size_t)(b * NQ + qi) * NN;
        const int half = lane >> 4;        // 0: lanes 0-15, 1: lanes 16-31
        const int l15  = lane & 15;
        // A-matrix row sources: row0 = y0, row1 = rho, rows 2..15 = 0
        const float* asrc = (l15 == 1) ? s_rho : s_y0;
        const bool   act  = (l15 < 2);

        float acc_uy = 0.f, acc_ur = 0.f, acc_wr = 0.f, acc_qy = 0.f, acc_qr = 0.f;

        for (int nt = 0; nt < 4; ++nt) {
            float u, w;  // u = (P y0)[col], w = (P rho)[col] on lanes 0..15
#ifdef HAVE_WMMA_F32
            v8f c = {};
            // B fragment base: col = nt*16 + l15, rows kb*4 + 2*half (+1)
            const float* Pb = Pq + half * (2 * NN) + nt * 16 + l15;
            #pragma unroll
            for (int kb = 0; kb < 16; ++kb) {
                int ko = 4 * kb + 2 * half;
                float ax = asrc[ko];
                float ay = asrc[ko + 1];
                ax = act ? ax : 0.0f;   // cndmask, EXEC stays full
                ay = act ? ay : 0.0f;
                v2f a;  a.x = ax;            a.y = ay;
                v2f bb; bb.x = Pb[kb * 4 * NN]; bb.y = Pb[kb * 4 * NN + NN];
                c = __builtin_amdgcn_wmma_f32_16x16x4_f32(
                        false, a, false, bb, (short)0, c, false, false);
            }
            u = c[0];   // C row M=0 on lanes 0..15
            w = c[1];   // C row M=1 on lanes 0..15
#else
            // VALU fallback: direct column dot products (coalesced by lane)
            u = 0.f; w = 0.f;
            const int col = nt * 16 + l15;
            #pragma unroll 8
            for (int k = 0; k < NN; ++k) {
                float pvv = Pq[k * NN + col];
                u += s_y0[k]  * pvv;
                w += s_rho[k] * pvv;
            }
#endif
            int   idx  = nt * 16 + l15;
            float qvl  = qq[idx];
            float y0v  = s_y0[idx];
            float rhv  = s_rho[idx];
            float msk  = (half == 0) ? 1.0f : 0.0f;  // rows M=8/9 on upper half: ignore
            acc_uy += msk * u * y0v;     // y0^T P y0 (via Py0 . y0)
            acc_ur += msk * u * rhv;     // (P y0) . rho
            acc_wr += msk * w * rhv;     // rho^T P rho
            acc_qy += msk * qvl * y0v;   // q . y0
            acc_qr += msk * qvl * rhv;   // q . rho
        }
        #pragma unroll
        for (int m = 16; m >= 1; m >>= 1) {
            acc_uy += __shfl_xor(acc_uy, m, 32);
            acc_ur += __shfl_xor(acc_ur, m, 32);
            acc_wr += __shfl_xor(acc_wr, m, 32);
            acc_qy += __shfl_xor(acc_qy, m, 32);
            acc_qr += __shfl_xor(acc_qr, m, 32);
        }
        if (lane == 0) {
            float rqv   = r[(size_t)b * NQ + qi];
            float c0    = 0.5f * acc_uy + acc_qy + rqv;   // < 0 strictly inside
            float sigma = 2.0f * c0;
            float gr    = acc_ur + acc_qr;                // g . rho
            float rPr   = acc_wr;
            float quad  = (gr * gr - 2.0f * c0 * rPr) / (sigma * sigma);
            float kq    = (-gr / sigma) + sqrtf(quad);
            s_red[8 + qi] = kq;
        }
    }
    __syncthreads();

    // ---------------- Phase D: combine, alpha, write output ---------------------
    if (tid == 0) {
        float k = 0.0f;
        #pragma unroll
        for (int i = 0; i < 16; ++i) k = fmaxf(k, s_red[i]);
        float nv    = s_scal[0];
        float alpha = fminf(1.0f / k, nv);   // k==0 -> +inf -> picks nv (matches ref)
        s_scal[1]   = alpha;
    }
    __syncthreads();
    if (tid < NN) {
        out[(size_t)b * NN + tid] =
            z0[(size_t)b * NN + tid] + s_scal[1] * s_rho[tid];
    }
}

extern "C" void kernel_launch(void* const* d_in, const int* in_sizes, int n_in,
                              void* d_out, int out_size, void* d_ws, size_t ws_size,
                              hipStream_t stream) {
    const float* v  = (const float*)d_in[0];
    const float* A1 = (const float*)d_in[1];
    const float* b1 = (const float*)d_in[2];
    const float* z0 = (const float*)d_in[3];
    const float* P  = (const float*)d_in[4];
    const float* qv = (const float*)d_in[5];
    const float* r  = (const float*)d_in[6];
    float* out = (float*)d_out;
    (void)in_sizes; (void)n_in; (void)out_size; (void)d_ws; (void)ws_size;
    constraint_kernel<<<NB, 256, 0, stream>>>(v, A1, b1, z0, P, qv, r, out);
}